// DirectPolicy_48601849922216
// MI455X (gfx1250) — compile-verified
//
#include <hip/hip_runtime.h>
#include <math.h>

#define BATCH      262144
#define DOUT       64
#define KY         16
#define HIDN       200
#define NOUTC      65
#define NEG_SLOPE  0.01f
#define ALPHA_C    0.3f
#define STRIDE     210          // even (float2-aligned) and 210%64=18 -> conflict-free A reads
#define WAVES      2
#define ROWS_W     16           // rows per wave (one 16xN WMMA M-tile)

// ---- packed-weight workspace layout (float offsets) ----
// W1p: [10][208][2]   (K=18 padded to 20 -> 10 k-pairs, N=200 padded to 208)
// W2p: [100][208][2]
// W3p: [100][80][2]   (N=65 padded to 80)
// b1p[208], b2p[208], b3p[80]
#define W1P_OFF   0
#define W1P_SZ    (10 * 208 * 2)      // 4160
#define W2P_OFF   (W1P_OFF + W1P_SZ)  // 4160
#define W2P_SZ    (100 * 208 * 2)     // 41600
#define W3P_OFF   (W2P_OFF + W2P_SZ)  // 45760
#define W3P_SZ    (100 * 80 * 2)      // 16000
#define B1P_OFF   (W3P_OFF + W3P_SZ)  // 61760
#define B2P_OFF   (B1P_OFF + 208)     // 61968
#define B3P_OFF   (B2P_OFF + 208)     // 62176
#define WS_FLOATS (B3P_OFF + 80)      // 62256
#define WS_BYTES  (WS_FLOATS * 4)     // 249,024 B

typedef __attribute__((ext_vector_type(2))) float v2f;
typedef __attribute__((ext_vector_type(8))) float v8f;

__device__ __forceinline__ float wave_sum(float x) {
#pragma unroll
    for (int o = 16; o > 0; o >>= 1) x += __shfl_xor(x, o, 32);
    return x;
}
__device__ __forceinline__ float wave_max(float x) {
#pragma unroll
    for (int o = 16; o > 0; o >>= 1) x = fmaxf(x, __shfl_xor(x, o, 32));
    return x;
}
// softplus(2x)/2, numerically stable
__device__ __forceinline__ float softplus2(float x) {
    float t = 2.0f * x;
    return 0.5f * (fmaxf(t, 0.0f) + log1pf(expf(-fabsf(t))));
}

// ---------------- weight prepack kernel (runs every launch; idempotent) ----------------
__global__ void prepack_kernel(const float* __restrict__ W1, const float* __restrict__ b1,
                               const float* __restrict__ W2, const float* __restrict__ b2,
                               const float* __restrict__ W3, const float* __restrict__ b3,
                               float* __restrict__ ws) {
    int i = blockIdx.x * blockDim.x + threadIdx.x;
    if (i >= WS_FLOATS) return;
    float v = 0.0f;
    if (i < W2P_OFF) {                       // W1p
        int idx = i - W1P_OFF;
        int j = idx & 1, rest = idx >> 1;
        int n = rest % 208, p = rest / 208;
        int k = 2 * p + j;
        if (k < 18 && n < 200) v = W1[k * 200 + n];
    } else if (i < W3P_OFF) {                // W2p
        int idx = i - W2P_OFF;
        int j = idx & 1, rest = idx >> 1;
        int n = rest % 208, p = rest / 208;
        int k = 2 * p + j;
        if (k < 200 && n < 200) v = W2[k * 200 + n];
    } else if (i < B1P_OFF) {                // W3p
        int idx = i - W3P_OFF;
        int j = idx & 1, rest = idx >> 1;
        int n = rest % 80, p = rest / 80;
        int k = 2 * p + j;
        if (k < 200 && n < 65) v = W3[k * 65 + n];
    } else if (i < B2P_OFF) {
        int n = i - B1P_OFF; if (n < 200) v = b1[n];
    } else if (i < B3P_OFF) {
        int n = i - B2P_OFF; if (n < 200) v = b2[n];
    } else {
        int n = i - B3P_OFF; if (n < 65) v = b3[n];
    }
    ws[i] = v;
}

// ---------------- packed-weight layer: no guards, b64 weight loads ----------------
template <int KSTEPS, int NT, bool ACT>
__device__ __forceinline__ void mlp_layer_packed(const float* __restrict__ Wp,   // [KSTEPS*2][NT*16][2]
                                                 const float* __restrict__ bp,   // [NT*16]
                                                 const float* in_lds, float* out_lds,
                                                 int laneLo, int half) {
    constexpr int NPAD = NT * 16;
    v8f acc[NT];
#pragma unroll
    for (int t = 0; t < NT; ++t) {
        float bv = bp[t * 16 + laneLo];
#pragma unroll
        for (int v = 0; v < 8; ++v) acc[t][v] = bv;
    }

#pragma unroll 2
    for (int ks = 0; ks < KSTEPS; ++ks) {
        int k0 = 4 * ks + 2 * half;   // A columns k0, k0+1 for this lane-half
        v2f a;
        a.x = in_lds[laneLo * STRIDE + k0];
        a.y = in_lds[laneLo * STRIDE + k0 + 1];
        int prow = 2 * ks + half;     // packed k-pair row
#pragma unroll
        for (int t = 0; t < NT; ++t) {
            v2f b = *(const v2f*)(Wp + (long)(prow * NPAD + t * 16 + laneLo) * 2);
            acc[t] = __builtin_amdgcn_wmma_f32_16x16x4_f32(
                false, a, false, b, (short)0, acc[t], false, false);
        }
    }

#pragma unroll
    for (int t = 0; t < NT; ++t) {
#pragma unroll
        for (int v = 0; v < 8; ++v) {
            float x = acc[t][v];
            if (ACT) x = (x > 0.0f) ? x : NEG_SLOPE * x;
            out_lds[(v + 8 * half) * STRIDE + t * 16 + laneLo] = x;
        }
    }
}

// ---------------- fallback layer (direct guarded loads from original weights) ----------------
template <int KSTEPS, int KROWS, int NT, int NCOLS, bool GUARDK, bool ACT>
__device__ __forceinline__ void mlp_layer_guarded(const float* __restrict__ W,
                                                  const float* __restrict__ bias,
                                                  const float* in_lds, float* out_lds,
                                                  int laneLo, int half) {
    v8f acc[NT];
#pragma unroll
    for (int t = 0; t < NT; ++t) {
        int nn = t * 16 + laneLo;
        float bv = (nn < NCOLS) ? bias[nn] : 0.0f;
#pragma unroll
        for (int v = 0; v < 8; ++v) acc[t][v] = bv;
    }
#pragma unroll 2
    for (int ks = 0; ks < KSTEPS; ++ks) {
        int k0 = 4 * ks + 2 * half;
        v2f a;
        a.x = in_lds[laneLo * STRIDE + k0];
        a.y = in_lds[laneLo * STRIDE + k0 + 1];
        // clamp addresses (always in-bounds) then select-zero: no exec-divergent loads
        int kc0 = k0, kc1 = k0 + 1;
        bool k0ok = true, k1ok = true;
        if (GUARDK) {
            k0ok = kc0 < KROWS; k1ok = kc1 < KROWS;
            kc0 = k0ok ? kc0 : KROWS - 1;
            kc1 = k1ok ? kc1 : KROWS - 1;
        }
#pragma unroll
        for (int t = 0; t < NT; ++t) {
            int nn = t * 16 + laneLo;
            bool nok = (nn < NCOLS);
            int nc = nok ? nn : NCOLS - 1;
            float w0 = W[(long)kc0 * NCOLS + nc];
            float w1 = W[(long)kc1 * NCOLS + nc];
            v2f b;
            b.x = (nok && k0ok) ? w0 : 0.0f;
            b.y = (nok && k1ok) ? w1 : 0.0f;
            acc[t] = __builtin_amdgcn_wmma_f32_16x16x4_f32(
                false, a, false, b, (short)0, acc[t], false, false);
        }
    }
#pragma unroll
    for (int t = 0; t < NT; ++t) {
#pragma unroll
        for (int v = 0; v < 8; ++v) {
            float x = acc[t][v];
            if (ACT) x = (x > 0.0f) ? x : NEG_SLOPE * x;
            out_lds[(v + 8 * half) * STRIDE + t * 16 + laneLo] = x;
        }
    }
}

// ---------------- shared device body pieces ----------------
__device__ __forceinline__ void load_z_tile(const float* __restrict__ X,
                                            const float* __restrict__ Y,
                                            const float* __restrict__ TmT,
                                            float* buf0, long base, int laneLo, int half) {
    if (half == 0) {
        long row = base + laneLo;
        float* zr = buf0 + laneLo * STRIDE;
        zr[0] = X[row];
#pragma unroll
        for (int k = 0; k < KY; ++k) zr[1 + k] = Y[row * KY + k];
        zr[17] = TmT[row];
        zr[18] = 0.0f;
        zr[19] = 0.0f;
    }
}

__device__ __forceinline__ void epilogue_tile(const float* __restrict__ X,
                                              const float* buf1, float* __restrict__ out,
                                              long base, int lane) {
    for (int r = 0; r < ROWS_W; ++r) {
        long row = base + r;
        const float* rr = buf1 + r * STRIDE;
        float u0 = softplus2(rr[2 * lane]);
        float u1 = softplus2(rr[2 * lane + 1]);   // softplus > 0 => vpos == u_pos

        float s = wave_sum(u0 + u1);
        float th = 0.0f;
        if (s > 1.0f) {   // wave-uniform branch
            float hi = wave_max(fmaxf(u0, u1));
            float lo = 0.0f;
#pragma unroll 1
            for (int it = 0; it < 28; ++it) {
                float mid = 0.5f * (lo + hi);
                float f = wave_sum(fmaxf(u0 - mid, 0.0f) + fmaxf(u1 - mid, 0.0f));
                if (f > 1.0f) lo = mid; else hi = mid;
            }
            th = 0.5f * (lo + hi);
        }
        long o = row * (long)NOUTC;
        out[o + 2 * lane]     = fmaxf(u0 - th, 0.0f);
        out[o + 2 * lane + 1] = fmaxf(u1 - th, 0.0f);
        if (lane == 0) {
            float v = rr[DOUT];
            out[o + DOUT] = ALPHA_C * X[row] * (1.0f / (1.0f + expf(-v)));
        }
    }
}

// ---------------- main kernels ----------------
__global__ __launch_bounds__(WAVES * 32)
void direct_policy_packed(const float* __restrict__ X, const float* __restrict__ Y,
                          const float* __restrict__ TmT, const float* __restrict__ ws,
                          float* __restrict__ out) {
    __shared__ float smem[WAVES * 2 * ROWS_W * STRIDE];  // 53.8 KB
    int tid = threadIdx.x, wave = tid >> 5, lane = tid & 31;
    int laneLo = lane & 15, half = lane >> 4;
    float* buf0 = smem + wave * 2 * ROWS_W * STRIDE;
    float* buf1 = buf0 + ROWS_W * STRIDE;
    long base = ((long)blockIdx.x * WAVES + wave) * ROWS_W;

    load_z_tile(X, Y, TmT, buf0, base, laneLo, half);
    // same-wave LDS ops are in-order; each wave owns its buffers -> no barrier needed
    mlp_layer_packed<5,  13, true >(ws + W1P_OFF, ws + B1P_OFF, buf0, buf1, laneLo, half);
    mlp_layer_packed<50, 13, true >(ws + W2P_OFF, ws + B2P_OFF, buf1, buf0, laneLo, half);
    mlp_layer_packed<50, 5,  false>(ws + W3P_OFF, ws + B3P_OFF, buf0, buf1, laneLo, half);
    epilogue_tile(X, buf1, out, base, lane);
}

__global__ __launch_bounds__(WAVES * 32)
void direct_policy_fallback(const float* __restrict__ X, const float* __restrict__ Y,
                            const float* __restrict__ TmT,
                            const float* __restrict__ W1, const float* __restrict__ b1,
                            const float* __restrict__ W2, const float* __restrict__ b2,
                            const float* __restrict__ W3, const float* __restrict__ b3,
                            float* __restrict__ out) {
    __shared__ float smem[WAVES * 2 * ROWS_W * STRIDE];
    int tid = threadIdx.x, wave = tid >> 5, lane = tid & 31;
    int laneLo = lane & 15, half = lane >> 4;
    float* buf0 = smem + wave * 2 * ROWS_W * STRIDE;
    float* buf1 = buf0 + ROWS_W * STRIDE;
    long base = ((long)blockIdx.x * WAVES + wave) * ROWS_W;

    load_z_tile(X, Y, TmT, buf0, base, laneLo, half);
    mlp_layer_guarded<5,  18,  13, HIDN,  true,  true >(W1, b1, buf0, buf1, laneLo, half);
    mlp_layer_guarded<50, 200, 13, HIDN,  false, true >(W2, b2, buf1, buf0, laneLo, half);
    mlp_layer_guarded<50, 200, 5,  NOUTC, false, false>(W3, b3, buf0, buf1, laneLo, half);
    epilogue_tile(X, buf1, out, base, lane);
}

extern "C" void kernel_launch(void* const* d_in, const int* in_sizes, int n_in,
                              void* d_out, int out_size, void* d_ws, size_t ws_size,
                              hipStream_t stream) {
    const float* X   = (const float*)d_in[0];
    const float* Y   = (const float*)d_in[1];
    const float* TmT = (const float*)d_in[2];
    const float* W1  = (const float*)d_in[3];
    const float* b1  = (const float*)d_in[4];
    const float* W2  = (const float*)d_in[5];
    const float* b2  = (const float*)d_in[6];
    const float* W3  = (const float*)d_in[7];
    const float* b3  = (const float*)d_in[8];
    float* out = (float*)d_out;

    dim3 grid(BATCH / (WAVES * ROWS_W));   // 8192 workgroups
    dim3 block(WAVES * 32);                // 2 waves of 32

    if (ws_size >= (size_t)WS_BYTES) {
        float* ws = (float*)d_ws;
        prepack_kernel<<<(WS_FLOATS + 255) / 256, 256, 0, stream>>>(W1, b1, W2, b2, W3, b3, ws);
        direct_policy_packed<<<grid, block, 0, stream>>>(X, Y, TmT, ws, out);
    } else {
        direct_policy_fallback<<<grid, block, 0, stream>>>(X, Y, TmT, W1, b1, W2, b2, W3, b3, out);
    }
}